// WindowAtt_16535624090241
// MI455X (gfx1250) — compile-verified
//
#include <hip/hip_runtime.h>

typedef __attribute__((ext_vector_type(2))) float v2f;
typedef __attribute__((ext_vector_type(8))) float v8f;

#define NHEAD 8
#define CHN   64
#define TT    12
#define VV    2048
#define NW    512
#define LL    48

// LDS row strides (floats); all even so v2f loads stay 8B-aligned
#define XS 66
#define OS 66
#define QS 34
#define AS 50

__device__ __forceinline__ v8f wmma4(v2f a, v2f b, v8f c) {
    // V_WMMA_F32_16X16X4_F32 : D = A(16x4) * B(4x16) + C(16x16)
    return __builtin_amdgcn_wmma_f32_16x16x4_f32(false, a, false, b, (short)0, c,
                                                 false, false);
}

__global__ __launch_bounds__(128, 1)
void winatt_fused(const float* __restrict__ x,
                  const float* __restrict__ w_qkv,
                  const float* __restrict__ b_qkv,
                  const float* __restrict__ w_proj,
                  const float* __restrict__ b_proj,
                  float* __restrict__ out)
{
    __shared__ __align__(16) float Xs[LL * XS];          // x slice, [l=t*4+j][ch]
    __shared__ __align__(16) float Os[LL * OS];          // attention output, [l][ch]
    __shared__ __align__(16) float WQ[32 * 8];           // w_qkv zero-padded to 32 rows
    __shared__ __align__(16) float BQ[32];               // b_qkv zero-padded
    __shared__ __align__(16) float QK[4 * LL * QS];      // per-wave qkv (48x32, cols 24..31 = 0)
    __shared__ __align__(16) float AT[4 * LL * AS];      // per-wave attention probs (48x48)

    const int tid  = threadIdx.x;
    const int lane = tid & 31;
    const int wave = tid >> 5;
    const int lo   = lane & 15;     // N / M index within a 16-tile
    const int hi   = lane >> 4;     // upper lane-half flag
    const int b    = blockIdx.x / NW;
    const int n    = blockIdx.x % NW;

    // ---- Phase 0: cooperative load of x[b, :, :, 4n..4n+3] into Xs[l][ch] ----
    const float* xb = x + ((size_t)b * CHN * TT) * VV + 4 * n;
    for (int i = tid; i < CHN * TT; i += 128) {
        int ch = i / TT, t = i % TT;
        float4 v = *(const float4*)(xb + ((size_t)ch * TT + t) * VV);
        int r0 = t * 4;
        Xs[(r0 + 0) * XS + ch] = v.x;
        Xs[(r0 + 1) * XS + ch] = v.y;
        Xs[(r0 + 2) * XS + ch] = v.z;
        Xs[(r0 + 3) * XS + ch] = v.w;
    }
    for (int i = tid; i < 256; i += 128) WQ[i] = (i < 24 * 8) ? w_qkv[i] : 0.0f;
    if (tid < 32) BQ[tid] = (tid < 24) ? b_qkv[tid] : 0.0f;
    __syncthreads();

    float* qk = QK + wave * (LL * QS);
    float* at = AT + wave * (LL * AS);
    const float scale = 0.35355339059327379f;  // 1/sqrt(8)

    // ---- Each wave handles 2 heads ----
    for (int hh = 0; hh < 2; ++hh) {
        const int h = wave * 2 + hh;

        // qkv(48x24, padded to 32) = Xs[:, h*8:+8] @ WQ^T + BQ
        for (int mt = 0; mt < 3; ++mt)
            for (int nt = 0; nt < 2; ++nt) {
                float bias = BQ[nt * 16 + lo];
                v8f acc = {bias, bias, bias, bias, bias, bias, bias, bias};
                for (int k = 0; k < 2; ++k) {
                    int col = k * 4 + 2 * hi;
                    v2f a  = *(const v2f*)&Xs[(mt * 16 + lo) * XS + h * 8 + col];
                    v2f bb = *(const v2f*)&WQ[(nt * 16 + lo) * 8 + col];
                    acc = wmma4(a, bb, acc);
                }
                int r0 = mt * 16 + hi * 8;
                for (int r = 0; r < 8; ++r)
                    qk[(r0 + r) * QS + nt * 16 + lo] = acc[r];
            }

        // S(48x48) = q @ k^T, scaled, into at[][]
        for (int mt = 0; mt < 3; ++mt)
            for (int nt = 0; nt < 3; ++nt) {
                v8f acc = {};
                for (int k = 0; k < 2; ++k) {
                    int col = k * 4 + 2 * hi;
                    v2f a  = *(const v2f*)&qk[(mt * 16 + lo) * QS + col];       // q
                    v2f bb = *(const v2f*)&qk[(nt * 16 + lo) * QS + 8 + col];  // k^T
                    acc = wmma4(a, bb, acc);
                }
                int r0 = mt * 16 + hi * 8;
                for (int r = 0; r < 8; ++r)
                    at[(r0 + r) * AS + nt * 16 + lo] = acc[r] * scale;
            }

        // row-wise softmax over 48 cols (wave-private buffer, DS ops in-order)
        {
            int nrows = (lane < 16) ? 2 : 1;
            for (int p = 0; p < nrows; ++p) {
                float* rp = at + (lane + p * 32) * AS;
                float m = -1e30f;
                for (int j = 0; j < LL; ++j) m = fmaxf(m, rp[j]);
                float s = 0.0f;
                for (int j = 0; j < LL; ++j) { float e = __expf(rp[j] - m); rp[j] = e; s += e; }
                float inv = 1.0f / s;
                for (int j = 0; j < LL; ++j) rp[j] *= inv;
            }
        }

        // O(48x8) = att @ v  (v = qkv cols 16..23; cols 24..31 are exact zeros)
        for (int mt = 0; mt < 3; ++mt) {
            v8f acc = {};
            for (int k = 0; k < 12; ++k) {
                int col = k * 4 + 2 * hi;
                v2f a = *(const v2f*)&at[(mt * 16 + lo) * AS + col];
                v2f bb;
                bb.x = qk[(col + 0) * QS + 16 + lo];
                bb.y = qk[(col + 1) * QS + 16 + lo];
                acc = wmma4(a, bb, acc);
            }
            if (lo < 8) {
                int r0 = mt * 16 + hi * 8;
                for (int r = 0; r < 8; ++r)
                    Os[(r0 + r) * OS + h * 8 + lo] = acc[r];
            }
        }
    }
    __syncthreads();

    // ---- Phase 3: out(48x64) = Os @ w_proj^T + b_proj; wave owns N-tile = wave ----
    {
        const int d = wave * 16 + lo;
        const float bias = b_proj[d];
        const float* wp = w_proj + (size_t)d * CHN;
        for (int mt = 0; mt < 3; ++mt) {
            v8f acc = {bias, bias, bias, bias, bias, bias, bias, bias};
            for (int k = 0; k < 16; ++k) {
                int col = k * 4 + 2 * hi;
                v2f a  = *(const v2f*)&Os[(mt * 16 + lo) * OS + col];
                v2f bb = *(const v2f*)(wp + col);
                acc = wmma4(a, bb, acc);
            }
            int l0 = mt * 16 + hi * 8;        // multiple of 4 -> j runs 0..3 in each float4
            int t0 = l0 >> 2;
            float* op = out + (((size_t)b * CHN + d) * TT + t0) * VV + 4 * n;
            *(float4*)op        = make_float4(acc[0], acc[1], acc[2], acc[3]);
            *(float4*)(op + VV) = make_float4(acc[4], acc[5], acc[6], acc[7]);
        }
    }
}

extern "C" void kernel_launch(void* const* d_in, const int* in_sizes, int n_in,
                              void* d_out, int out_size, void* d_ws, size_t ws_size,
                              hipStream_t stream) {
    const float* x      = (const float*)d_in[0];
    const float* w_qkv  = (const float*)d_in[1];
    const float* b_qkv  = (const float*)d_in[2];
    const float* w_proj = (const float*)d_in[3];
    const float* b_proj = (const float*)d_in[4];
    float* out = (float*)d_out;

    dim3 grid(16 * NW);   // one workgroup per (batch, window)
    dim3 block(128);      // 4 wave32s
    hipLaunchKernelGGL(winatt_fused, grid, block, 0, stream,
                       x, w_qkv, b_qkv, w_proj, b_proj, out);
}